// LightGCN_60035052863927
// MI455X (gfx1250) — compile-verified
//
#include <hip/hip_runtime.h>
#include <hip/hip_bf16.h>

// LightGCN propagation for MI455X (gfx1250, wave32).
// N = 150000 nodes, D = 64, NNZ = 2,000,000, 3 layers.
// Memory-bound: dense state (38.4 MB/buffer) is L2-resident (192 MB L2).
// SpMM via fp32 L2 atomics; row L2-norms via fp32 WMMA (Gram diagonal).

#define NUM_USERS 100000
#define NUM_ITEMS 50000
#define NNODES    (NUM_USERS + NUM_ITEMS)   // 150000
#define DIM       64
#define NNZ_E     2000000
#define NTILES    (NNODES / 16)             // 9375
#define NORM_BLOCKS (NTILES / 5)            // 1875 blocks x 5 waves
#define VEC4N     ((size_t)NNODES * DIM / 4) // 2,400,000 float4
#define USER_V4   ((size_t)NUM_USERS * DIM / 4) // 1,600,000

typedef float v2f __attribute__((ext_vector_type(2)));
typedef float v4f __attribute__((ext_vector_type(4)));
typedef float v8f __attribute__((ext_vector_type(8)));

// ---------------------------------------------------------------------------
// Kernel 1: concat(user, item) -> layer_sum (d_out) and h0
// One float4 per thread; grid sized exactly (9375 x 256 = 2.4M threads).
// ---------------------------------------------------------------------------
__global__ __launch_bounds__(256) void lg_init(const float* __restrict__ eu,
                                               const float* __restrict__ ei,
                                               float* __restrict__ out,
                                               float* __restrict__ h0) {
    size_t i = (size_t)blockIdx.x * 256 + threadIdx.x;   // float4 index
    v4f v = (i < USER_V4) ? ((const v4f*)eu)[i]
                          : ((const v4f*)ei)[i - USER_V4];
    ((v4f*)out)[i] = v;
    ((v4f*)h0)[i]  = v;
}

// ---------------------------------------------------------------------------
// Kernel 2: COO SpMM scatter: y[row,:] += val * x[col,:]
// One edge per wave (grid = NNZ/8 blocks x 8 waves, exact).
// Edge index forced scalar -> s_load of rows/cols/vals; the 256 B row gather
// is fully coalesced across the 32 lanes (float2 each); 2 fp32 atomics/lane
// hit the L2 atomic units (data set is L2-resident).
// ---------------------------------------------------------------------------
__global__ __launch_bounds__(256) void lg_spmm(const int* __restrict__ rows,
                                               const int* __restrict__ cols,
                                               const float* __restrict__ vals,
                                               const float* __restrict__ x,
                                               float* __restrict__ y) {
    int lane = threadIdx.x & 31;
    int e = __builtin_amdgcn_readfirstlane((int)(blockIdx.x * 8 + (threadIdx.x >> 5)));
    int r = rows[e];
    int c = cols[e];
    float v = vals[e];
    v2f xv = *(const v2f*)(x + (size_t)c * DIM + lane * 2);
    float* yp = y + (size_t)r * DIM + lane * 2;
    atomicAdd(yp,     v * xv.x);
    atomicAdd(yp + 1, v * xv.y);
}

// ---------------------------------------------------------------------------
// Kernel 3: per-row L2 normalize + accumulate, one 16-row tile per wave.
// Row norms^2 = diagonal of G = H_tile (16x64) * H_tile^T, computed with a
// chain of 16 fully-fp32 V_WMMA_F32_16X16X4_F32 ops (K=4 each).
// A-matrix 16x4 f32 layout (ISA 7.12.2): lane L holds row M=L%16, VGPR j
// holds K = 2*(L>>4)+j.  The row-striped B layout for the transposed chunk
// yields identical per-lane contents, so A and B share registers.
// D layout: VGPR i = row M=i (lanes 0-15) / M=i+8 (lanes 16-31), N = lane%16.
// Diagonal owners: lanes 0..7 -> rows 0..7 (c[lane]); lanes 24..31 ->
// rows 8..15 (c[lane-24]).  Broadcast via __shfl, then scaled RMW into sum.
// Grid = 1875 blocks x 5 waves (exact) -> EXEC always all-ones for WMMA.
// ---------------------------------------------------------------------------
__global__ __launch_bounds__(160) void lg_norm_accum(const float* __restrict__ h,
                                                     float* __restrict__ sum) {
    int lane = threadIdx.x & 31;
    int wave = __builtin_amdgcn_readfirstlane((int)(threadIdx.x >> 5));
    int tile = blockIdx.x * 5 + wave;                 // < 9375 always
    int m    = lane & 15;                             // row within tile
    int half = lane >> 4;                             // K-pair selector

    const float* hp = h + (size_t)(tile * 16 + m) * DIM + half * 2;
    v2f a[16];
#pragma unroll
    for (int q = 0; q < 16; ++q)
        a[q] = *(const v2f*)(hp + q * 4);

    v8f c = {};
#pragma unroll
    for (int q = 0; q < 16; ++q)
        c = __builtin_amdgcn_wmma_f32_16x16x4_f32(
                /*neg_a=*/false, a[q], /*neg_b=*/false, a[q],
                /*c_mod=*/(short)0, c, /*reuse_a=*/false, /*reuse_b=*/false);

    float g = c[lane & 7];                            // this lane's diagonal entry
    int src = (m < 8) ? m : (m + 8);
    float nsq = __shfl(g, src, 32);
    float s = 1.0f / fmaxf(sqrtf(nsq), 1e-12f);       // h / max(||h||, eps)

    float* sp = sum + (size_t)(tile * 16 + m) * DIM + half * 2;
#pragma unroll
    for (int q = 0; q < 16; ++q) {
        v2f t = *(const v2f*)(sp + q * 4);
        t.x += a[q].x * s;
        t.y += a[q].y * s;
        *(v2f*)(sp + q * 4) = t;
    }
}

// ---------------------------------------------------------------------------
// Kernel 4: mean over (NUM_LAYERS+1) stacked embeddings: out *= 0.25
// ---------------------------------------------------------------------------
__global__ __launch_bounds__(256) void lg_scale(float* __restrict__ out) {
    size_t i = (size_t)blockIdx.x * 256 + threadIdx.x;
    v4f v = ((const v4f*)out)[i];
    v *= 0.25f;
    ((v4f*)out)[i] = v;
}

extern "C" void kernel_launch(void* const* d_in, const int* in_sizes, int n_in,
                              void* d_out, int out_size, void* d_ws, size_t ws_size,
                              hipStream_t stream) {
    const float* eu   = (const float*)d_in[0];   // [100000,64] f32
    const float* ei   = (const float*)d_in[1];   // [50000,64]  f32
    const int*   rows = (const int*)  d_in[2];   // [2M] i32
    const int*   cols = (const int*)  d_in[3];   // [2M] i32
    const float* vals = (const float*)d_in[4];   // [2M] f32
    float* out = (float*)d_out;                  // [150000,64] f32 (users||items)

    const size_t nodeBytes = (size_t)NNODES * DIM * sizeof(float);  // 38.4 MB
    float* h0 = (float*)d_ws;
    float* h1 = (float*)((char*)d_ws + nodeBytes);

    // layer_sum = all_embed (into d_out); h0 = all_embed
    lg_init<<<(int)(VEC4N / 256), 256, 0, stream>>>(eu, ei, out, h0);

    float* cur = h0;
    float* nxt = h1;
    for (int layer = 0; layer < 3; ++layer) {
        hipMemsetAsync(nxt, 0, nodeBytes, stream);
        lg_spmm<<<NNZ_E / 8, 256, 0, stream>>>(rows, cols, vals, cur, nxt);
        lg_norm_accum<<<NORM_BLOCKS, 160, 0, stream>>>(nxt, out);
        float* t = cur; cur = nxt; nxt = t;   // next layer propagates un-normalized h
    }

    lg_scale<<<(int)(VEC4N / 256), 256, 0, stream>>>(out);
}